// GPUAugment_33517924777993
// MI455X (gfx1250) — compile-verified
//
#include <hip/hip_runtime.h>
#include <stdint.h>

typedef unsigned int u32x4 __attribute__((ext_vector_type(4)));
typedef int          i32x4 __attribute__((ext_vector_type(4)));
typedef int          i32x8 __attribute__((ext_vector_type(8)));
typedef float        f32x4 __attribute__((ext_vector_type(4)));

#define HW      64
#define ROWF    192            // floats per image row (64 px * 3 ch)
#define IMGF    12288          // floats per image (64*64*3)
#define NPIX    4096           // pixels per image
#define THREADS 256

// One workgroup per image. LDS: A = source image (48KB, later reused as blur
// temp), B = working image (48KB), R = 1-float reduction slot.
extern "C" __global__ void __launch_bounds__(THREADS)
augment_kernel(const float* __restrict__ x,
               const float* __restrict__ crop,
               const unsigned char* __restrict__ flipm,
               const float* __restrict__ bright,
               const float* __restrict__ contr,
               const unsigned char* __restrict__ blurm,
               const float* __restrict__ sig,
               float* __restrict__ out)
{
    extern __shared__ float smem[];
    float* A = smem;               // [0, IMGF)
    float* B = smem + IMGF;        // [IMGF, 2*IMGF)
    float* R = smem + 2 * IMGF;    // reduction slot

    const int b = blockIdx.x;
    const int t = threadIdx.x;

    // ---- TDM: DMA the whole 64x64x3 fp32 image into LDS (wave 0 only) ----
    if (t < 32) {
        if (t == 0) R[0] = 0.0f;
        const uint64_t ga  = (uint64_t)(uintptr_t)(x + (size_t)b * IMGF);
        const uint32_t lds = (uint32_t)(uintptr_t)A;   // LDS byte offset = addr[31:0]
        u32x4 g0;
        g0[0] = 1u;                                    // count=1, user descriptor
        g0[1] = lds;                                   // lds_addr
        g0[2] = (uint32_t)(ga & 0xffffffffu);          // global_addr[31:0]
        g0[3] = (uint32_t)((ga >> 32) & 0x01ffffffu)   // global_addr[56:32]
              | (2u << 30);                            // type = 2 ("image")
        i32x8 g1;
        g1[0] = (int)(2u << 16);       // data_size = 4 bytes
        g1[1] = (int)(192u << 16);     // tensor_dim0 = 192  (bits [63:48])
        g1[2] = (int)(64u  << 16);     // tensor_dim1 = 64   (bits [95:80])
        g1[3] = (int)(192u << 16);     // tile_dim0   = 192  (bits [127:112])
        g1[4] = 64;                    // tile_dim1   = 64
        g1[5] = 192;                   // tensor_dim0_stride = 192
        g1[6] = 0;
        g1[7] = 0;
        i32x4 gz = {0, 0, 0, 0};       // 2D tensor: groups 2/3 unused
#if defined(__clang_major__) && (__clang_major__ >= 23)
        i32x8 gz8 = {0, 0, 0, 0, 0, 0, 0, 0};
        __builtin_amdgcn_tensor_load_to_lds(g0, g1, gz, gz, gz8, 0);
#else
        __builtin_amdgcn_tensor_load_to_lds(g0, g1, gz, gz, 0);
#endif
        __builtin_amdgcn_s_wait_tensorcnt(0);
    }
    __syncthreads();   // image resident in LDS A; R initialized

    // ---- per-image (block-uniform) parameters ----
    const float cy0 = crop[b * 4 + 0];
    const float cx0 = crop[b * 4 + 1];
    const float chh = crop[b * 4 + 2];
    const float cww = crop[b * 4 + 3];
    const bool  flip = flipm[b] != 0;
    const float br   = bright[b];
    const float cf   = contr[b];
    const bool  dob  = blurm[b] != 0;
    const float sg   = sig[b];

    // ---- bilinear crop-resize (+flip folded into x) + brightness + gray sum ----
    float gsum = 0.0f;
    for (int p = t; p < NPIX; p += THREADS) {
        const int y  = p >> 6;
        const int xo = p & 63;
        const int xe = flip ? (63 - xo) : xo;
        const float sy = cy0 + (y  + 0.5f) * chh * (1.0f / 64.0f) - 0.5f;
        const float sx = cx0 + (xe + 0.5f) * cww * (1.0f / 64.0f) - 0.5f;
        const float fy = floorf(sy), fx = floorf(sx);
        const float wy = sy - fy,    wx = sx - fx;
        int yi0 = (int)fy; yi0 = yi0 < 0 ? 0 : (yi0 > 63 ? 63 : yi0);
        int xi0 = (int)fx; xi0 = xi0 < 0 ? 0 : (xi0 > 63 ? 63 : xi0);
        const int yi1 = (yi0 + 1 > 63) ? 63 : yi0 + 1;
        const int xi1 = (xi0 + 1 > 63) ? 63 : xi0 + 1;
        const float* r0 = A + yi0 * ROWF;
        const float* r1 = A + yi1 * ROWF;
        const float w00 = (1.0f - wy) * (1.0f - wx);
        const float w01 = (1.0f - wy) * wx;
        const float w10 = wy * (1.0f - wx);
        const float w11 = wy * wx;
        float pix[3];
#pragma unroll
        for (int c = 0; c < 3; ++c) {
            float v = w00 * r0[xi0 * 3 + c] + w01 * r0[xi1 * 3 + c]
                    + w10 * r1[xi0 * 3 + c] + w11 * r1[xi1 * 3 + c];
            v = v * br;
            v = v < 0.0f ? 0.0f : (v > 1.0f ? 1.0f : v);
            pix[c] = v;
            B[p * 3 + c] = v;
        }
        gsum += 0.2989f * pix[0] + 0.587f * pix[1] + 0.114f * pix[2];
    }

    // ---- grayscale mean: wave32 shuffle tree + one LDS atomic per wave ----
#pragma unroll
    for (int o = 16; o > 0; o >>= 1) gsum += __shfl_xor(gsum, o, 32);
    if ((t & 31) == 0) atomicAdd(R, gsum);
    __syncthreads();
    const float m = R[0] * (1.0f / (float)NPIX);

    // ---- contrast blend + clip (in place on B) ----
    for (int i = t; i < IMGF; i += THREADS) {
        float v = cf * B[i] + (1.0f - cf) * m;
        B[i] = v < 0.0f ? 0.0f : (v > 1.0f ? 1.0f : v);
    }
    __syncthreads();

    // ---- conditional separable 3-tap blur (block-uniform branch) ----
    if (dob) {
        const float e  = __expf(-1.0f / (2.0f * sg * sg));
        const float k1 = 1.0f / (1.0f + 2.0f * e);
        const float k0 = e * k1;                    // symmetric: k0 == k2
        // vertical (reflect pad): B -> A
        for (int i = t; i < IMGF; i += THREADS) {
            const int y  = (int)((unsigned)i / ROWF);
            const int r  = i - y * ROWF;
            const int ym = (y == 0)  ? 1  : y - 1;
            const int yp = (y == 63) ? 62 : y + 1;
            A[i] = k0 * (B[ym * ROWF + r] + B[yp * ROWF + r]) + k1 * B[i];
        }
        __syncthreads();
        // horizontal (reflect pad): A -> B
        for (int i = t; i < IMGF; i += THREADS) {
            const int y  = (int)((unsigned)i / ROWF);
            const int r  = i - y * ROWF;
            const int xx = (int)((unsigned)r / 3u);
            const int c  = r - xx * 3;
            const int xm = (xx == 0)  ? 1  : xx - 1;
            const int xp = (xx == 63) ? 62 : xx + 1;
            B[i] = k0 * (A[y * ROWF + xm * 3 + c] + A[y * ROWF + xp * 3 + c])
                 + k1 * A[i];
        }
        __syncthreads();
    }

    // ---- normalize + HWC->CHW transpose, float4 non-temporal stores ----
    f32x4* outv = (f32x4*)(out + (size_t)b * IMGF);
    for (int j = t; j < IMGF / 4; j += THREADS) {
        const int o   = j << 2;          // element offset in CHW image
        const int c   = o >> 12;         // / 4096
        const int rem = o & 4095;
        const int y   = rem >> 6;
        const int xb  = rem & 63;
        const float mn = (c == 0) ? 0.485f : ((c == 1) ? 0.456f : 0.406f);
        const float is = (c == 0) ? (1.0f / 0.229f)
                        : ((c == 1) ? (1.0f / 0.224f) : (1.0f / 0.225f));
        const float* row = B + y * ROWF + xb * 3 + c;   // stride-3: bank-conflict-free
        f32x4 v;
        v.x = (row[0] - mn) * is;
        v.y = (row[3] - mn) * is;
        v.z = (row[6] - mn) * is;
        v.w = (row[9] - mn) * is;
        __builtin_nontemporal_store(v, outv + j);
    }
}

extern "C" void kernel_launch(void* const* d_in, const int* in_sizes, int n_in,
                              void* d_out, int out_size, void* d_ws, size_t ws_size,
                              hipStream_t stream) {
    (void)n_in; (void)out_size; (void)d_ws; (void)ws_size;
    const float*         xin    = (const float*)d_in[0];
    const float*         cropv  = (const float*)d_in[1];
    const unsigned char* flipv  = (const unsigned char*)d_in[2];
    const float*         brightv= (const float*)d_in[3];
    const float*         contrv = (const float*)d_in[4];
    const unsigned char* blurv  = (const unsigned char*)d_in[5];
    const float*         sigv   = (const float*)d_in[6];
    float*               outp   = (float*)d_out;

    const int Bn = in_sizes[3];                       // 4096 images
    const size_t shmem = (size_t)(2 * IMGF + 16) * sizeof(float);  // ~96 KB

    // Opt in to >64KB dynamic LDS (CDNA5 WGP has 320KB).
    (void)hipFuncSetAttribute((const void*)augment_kernel,
                              hipFuncAttributeMaxDynamicSharedMemorySize,
                              (int)shmem);

    augment_kernel<<<Bn, THREADS, shmem, stream>>>(
        xin, cropv, flipv, brightv, contrv, blurv, sigv, outp);
}